// GCN_27530740367364
// MI455X (gfx1250) — compile-verified
//
#include <hip/hip_runtime.h>
#include <math.h>

// ---- problem constants (match reference) ----
#define N_NODES  100000
#define N_EDGES  1000000
#define D_FEAT   64
#define D_EDGE   16
#define EMB      64
#define NODE_TILES (N_NODES / 16)   // 6250, exact
#define EDGE_TILES (N_EDGES / 16)   // 62500, exact
#define WPB 8                       // waves per block (256 threads, wave32)

// LDS layout for B-operand weights: k-pair interleaved.
//   (k, c) -> sW[(k/2)*PITCH + 2*c + (k&1)]
// so the WMMA B fragment {K=kb, K=kb+1} at column c is one aligned 8-byte
// word -> single ds_load_b64 into an even VGPR pair (no shuffle movs).
// PITCH = 160 dwords: pair-row p+1 is +160 = +32 (mod 64 banks) vs pair-row p,
// so lanes 0-15 (row p) use banks 0-31 and lanes 16-31 (row p+1) use banks
// 32-63 -> conflict-free wave32 b64 LDS reads.
#define PITCH 160

typedef __attribute__((ext_vector_type(2))) float v2f;
typedef __attribute__((ext_vector_type(8))) float v8f;

// ---------------------------------------------------------------------------
// Kernel 1: degree counts (senders / receivers), int atomics into workspace
// ---------------------------------------------------------------------------
__global__ __launch_bounds__(256) void gcn_degrees(
    const int* __restrict__ senders, const int* __restrict__ receivers,
    int* __restrict__ deg_s, int* __restrict__ deg_r)
{
    int e = blockIdx.x * 256 + threadIdx.x;
    if (e < N_EDGES) {
        atomicAdd(&deg_s[senders[e]], 1);
        atomicAdd(&deg_r[receivers[e]], 1);
    }
}

// ---------------------------------------------------------------------------
// Kernel 2: node projection  proj = nodes @ W + Wb   via V_WMMA_F32_16X16X4_F32
// One wave handles a 16-row tile x 64 cols (4 accumulators of 16x16).
// Fragment layouts (ISA 7.12.2):
//   A (16x4 f32):  lane l, vgpr v -> M = l%16, K = v + 2*(l/16)
//   B (4x16 f32):  lane l, vgpr v -> K = v + 2*(l/16), N = l%16
//   C/D (16x16):   lane l, vgpr v -> M = v + 8*(l/16), N = l%16
// ---------------------------------------------------------------------------
__global__ __launch_bounds__(256) void gcn_node_proj(
    const float* __restrict__ nodes, const float* __restrict__ W,
    const float* __restrict__ Wb, float* __restrict__ proj)
{
    __shared__ float sW[(D_FEAT / 2) * PITCH];      // 32 pair-rows, 20 KB
    for (int i = threadIdx.x; i < D_FEAT * EMB; i += 256) {
        const int k = i >> 6, c = i & 63;
        sW[(k >> 1) * PITCH + 2 * c + (k & 1)] = W[i];
    }
    __syncthreads();

    const int wave = threadIdx.x >> 5;
    const int lane = threadIdx.x & 31;
    const int tile = blockIdx.x * WPB + wave;       // wave-uniform -> full EXEC
    if (tile >= NODE_TILES) return;

    const int hi = lane >> 4;                       // 0 or 1
    const int lo = lane & 15;
    const int m0 = tile * 16;

    const float* arow = nodes + (size_t)(m0 + lo) * D_FEAT;

    v8f acc[4];
    #pragma unroll
    for (int nt = 0; nt < 4; ++nt) acc[nt] = (v8f){0,0,0,0,0,0,0,0};

    #pragma unroll
    for (int kk = 0; kk < D_FEAT / 4; ++kk) {       // 16 k-steps of K=4
        const int kb = kk * 4 + 2 * hi;             // even -> aligned b64
        const v2f a = *(const v2f*)&arow[kb];
        const float* bp = &sW[(2 * kk + hi) * PITCH + 2 * lo];
        #pragma unroll
        for (int nt = 0; nt < 4; ++nt) {
            const v2f b = *(const v2f*)&bp[nt * 32];
            acc[nt] = __builtin_amdgcn_wmma_f32_16x16x4_f32(
                false, a, false, b, (short)0, acc[nt], false, false);
        }
    }

    #pragma unroll
    for (int nt = 0; nt < 4; ++nt) {
        const int col = nt * 16 + lo;
        const float bias = Wb[col];
        #pragma unroll
        for (int v = 0; v < 8; ++v) {
            const int r = m0 + v + 8 * hi;
            proj[(size_t)r * EMB + col] = acc[nt][v] + bias;
        }
    }
}

// ---------------------------------------------------------------------------
// Kernel 3: per-edge message + scatter.
// C accumulator is pre-loaded with gathered proj[senders] + We_bias, then
// WMMA accumulates edges @ We on top. Results scatter-added to out[receivers]
// with global_atomic_add_f32, using the C/D fragment layout for row indexing.
// ---------------------------------------------------------------------------
__global__ __launch_bounds__(256) void gcn_edge_msg(
    const float* __restrict__ edges, const float* __restrict__ We,
    const float* __restrict__ Web, const float* __restrict__ proj,
    const int* __restrict__ senders, const int* __restrict__ receivers,
    float* __restrict__ out)
{
    __shared__ float sWe[(D_EDGE / 2) * PITCH];     // 8 pair-rows, 5 KB
    for (int i = threadIdx.x; i < D_EDGE * EMB; i += 256) {
        const int k = i >> 6, c = i & 63;
        sWe[(k >> 1) * PITCH + 2 * c + (k & 1)] = We[i];
    }
    __syncthreads();

    const int wave = threadIdx.x >> 5;
    const int lane = threadIdx.x & 31;
    const int tile = blockIdx.x * WPB + wave;       // wave-uniform
    if (tile >= EDGE_TILES) return;

    const int hi = lane >> 4;
    const int lo = lane & 15;
    const int e0 = tile * 16;

    // C init: gathered sender projection (L2-resident table) + edge bias
    v8f acc[4];
    float bias[4];
    #pragma unroll
    for (int nt = 0; nt < 4; ++nt) bias[nt] = Web[nt * 16 + lo];
    #pragma unroll
    for (int v = 0; v < 8; ++v) {
        const int r = v + 8 * hi;
        const int s = senders[e0 + r];
        const float* prow = proj + (size_t)s * EMB;
        #pragma unroll
        for (int nt = 0; nt < 4; ++nt)
            acc[nt][v] = prow[nt * 16 + lo] + bias[nt];
    }

    const float* erow = edges + (size_t)(e0 + lo) * D_EDGE;

    #pragma unroll
    for (int kk = 0; kk < D_EDGE / 4; ++kk) {       // 4 k-steps of K=4
        const int kb = kk * 4 + 2 * hi;
        const v2f a = *(const v2f*)&erow[kb];
        const float* bp = &sWe[(2 * kk + hi) * PITCH + 2 * lo];
        #pragma unroll
        for (int nt = 0; nt < 4; ++nt) {
            const v2f b = *(const v2f*)&bp[nt * 32];
            acc[nt] = __builtin_amdgcn_wmma_f32_16x16x4_f32(
                false, a, false, b, (short)0, acc[nt], false, false);
        }
    }

    // scatter-add into out[receiver] (f32 atomics, L2-resident target)
    #pragma unroll
    for (int v = 0; v < 8; ++v) {
        const int r   = v + 8 * hi;
        const int rec = receivers[e0 + r];
        float* orow = out + (size_t)rec * EMB;
        #pragma unroll
        for (int nt = 0; nt < 4; ++nt)
            atomicAdd(&orow[nt * 16 + lo], acc[nt][v]);
    }
}

// ---------------------------------------------------------------------------
// Kernel 4: symmetric degree normalization, in place.
// ---------------------------------------------------------------------------
__global__ __launch_bounds__(256) void gcn_norm(
    float* __restrict__ out, const int* __restrict__ deg_s,
    const int* __restrict__ deg_r)
{
    int idx = blockIdx.x * 256 + threadIdx.x;
    if (idx < N_NODES * EMB) {
        const int n = idx >> 6;
        const int ds = max(deg_s[n], 1);
        const int dr = max(deg_r[n], 1);
        out[idx] = out[idx] / sqrtf((float)(ds * dr));
    }
}

// ---------------------------------------------------------------------------
extern "C" void kernel_launch(void* const* d_in, const int* in_sizes, int n_in,
                              void* d_out, int out_size, void* d_ws, size_t ws_size,
                              hipStream_t stream) {
    (void)in_sizes; (void)n_in; (void)out_size; (void)ws_size;
    const float* nodes     = (const float*)d_in[0];
    const float* edges     = (const float*)d_in[1];
    const int*   senders   = (const int*)d_in[2];
    const int*   receivers = (const int*)d_in[3];
    const float* Wk        = (const float*)d_in[4];
    const float* Wb        = (const float*)d_in[5];
    const float* Wek       = (const float*)d_in[6];
    const float* Web       = (const float*)d_in[7];
    float* out = (float*)d_out;

    // workspace layout: proj (25.6 MB) | deg_s (400 KB) | deg_r (400 KB)
    char*  ws    = (char*)d_ws;
    float* proj  = (float*)ws;
    int*   deg_s = (int*)(ws + (size_t)N_NODES * EMB * sizeof(float));
    int*   deg_r = deg_s + N_NODES;

    // zero accumulation targets every call (graph-capturable stream memsets)
    hipMemsetAsync(out,   0, (size_t)N_NODES * EMB * sizeof(float), stream);
    hipMemsetAsync(deg_s, 0, 2 * (size_t)N_NODES * sizeof(int), stream);

    gcn_degrees<<<(N_EDGES + 255) / 256, 256, 0, stream>>>(senders, receivers, deg_s, deg_r);
    gcn_node_proj<<<(NODE_TILES + WPB - 1) / WPB, 256, 0, stream>>>(nodes, Wk, Wb, proj);
    gcn_edge_msg<<<(EDGE_TILES + WPB - 1) / WPB, 256, 0, stream>>>(
        edges, Wek, Web, proj, senders, receivers, out);
    gcn_norm<<<(N_NODES * EMB + 255) / 256, 256, 0, stream>>>(out, deg_s, deg_r);
}